// ColorCurveLearningLoss_16312285790272
// MI455X (gfx1250) — compile-verified
//
#include <hip/hip_runtime.h>
#include <hip/hip_bf16.h>

// ---------------------------------------------------------------------------
// ColorCurveLearningLoss for MI455X (gfx1250, wave32)
//
// Memory-bound: 3 x 50MB f32 streams -> ~6.5us at 23.3 TB/s. Strategy:
//   * float4 (b128) non-temporal global loads, grid-stride
//   * per-WAVE privatized LDS histograms (96 segs x {sumP,sumT,cnt}) via
//     ds_add_f32 atomics  (8 waves x 288 floats = 9KB of the 320KB WGP LDS)
//   * one global_atomic_add_f32 merge per block per segment into d_ws
//   * tiny finalize kernel: per-bin means, |diff|, exact-f32 total via
//     V_WMMA_F32_16X16X4_F32 (A = diffs, B = ones -> row sums) + shfl_xor(16)
// ---------------------------------------------------------------------------

#define NBINS   32
#define NSEG    96          // 3 channels * 32 bins
#define WAVES   8           // 256-thread block = 8 wave32
#define VEC_PER_PLANE_SHIFT 16   // (512*512)/4 = 65536 float4 per HW plane

typedef float f4  __attribute__((ext_vector_type(4)));
typedef float v2f __attribute__((ext_vector_type(2)));
typedef float v8f __attribute__((ext_vector_type(8)));

// ws layout: [0..95]=sum_pred  [96..191]=sum_target  [192..287]=count
__global__ void ccl_zero_ws(float* __restrict__ ws) {
    for (int i = threadIdx.x; i < 3 * NSEG; i += blockDim.x) ws[i] = 0.0f;
}

__global__ __launch_bounds__(256)
void ccl_hist(const f4* __restrict__ pred,
              const f4* __restrict__ targ,
              const f4* __restrict__ img,
              float* __restrict__ ws,
              int nvec) {
    __shared__ float lhist[WAVES * 3 * NSEG];

    for (int i = threadIdx.x; i < WAVES * 3 * NSEG; i += blockDim.x)
        lhist[i] = 0.0f;
    __syncthreads();

    const int wave = threadIdx.x >> 5;
    float* wsum_p = &lhist[wave * 3 * NSEG + 0 * NSEG];
    float* wsum_t = &lhist[wave * 3 * NSEG + 1 * NSEG];
    float* wcnt   = &lhist[wave * 3 * NSEG + 2 * NSEG];

    const int stride = gridDim.x * blockDim.x;
    for (int v = blockIdx.x * blockDim.x + threadIdx.x; v < nvec; v += stride) {
        // prefetch next grid-stride iteration (speculative; OOB is dropped)
        __builtin_prefetch(&img[v + stride],  0, 0);
        __builtin_prefetch(&pred[v + stride], 0, 0);
        __builtin_prefetch(&targ[v + stride], 0, 0);

        // channel: flat plane index = v / 65536, c = plane % 3 (NCHW layout)
        const int cbase = ((v >> VEC_PER_PLANE_SHIFT) % 3) * NBINS;

        const f4 x = __builtin_nontemporal_load(&img[v]);
        const f4 p = __builtin_nontemporal_load(&pred[v]);
        const f4 t = __builtin_nontemporal_load(&targ[v]);

        // edges are exact multiples of 2^-5 and x*32 is an exact fp32 multiply,
        // so (int)(x*32) == searchsorted(edges, x, 'right') - 1 for x in [0,1)
#define CCL_ELEM(XX, PP, TT)                                        \
        do {                                                        \
            const float xe = (XX);                                  \
            if (xe >= 0.0f && xe < 1.0f) {                          \
                const int b = cbase + (int)(xe * 32.0f);            \
                atomicAdd(&wsum_p[b], (PP));                        \
                atomicAdd(&wsum_t[b], (TT));                        \
                atomicAdd(&wcnt[b], 1.0f);                          \
            }                                                       \
        } while (0)

        CCL_ELEM(x.x, p.x, t.x);
        CCL_ELEM(x.y, p.y, t.y);
        CCL_ELEM(x.z, p.z, t.z);
        CCL_ELEM(x.w, p.w, t.w);
#undef CCL_ELEM
    }
    __syncthreads();

    // fold the 8 per-wave histograms, one global atomic per segment per block
    for (int i = threadIdx.x; i < 3 * NSEG; i += blockDim.x) {
        float s = 0.0f;
#pragma unroll
        for (int w = 0; w < WAVES; ++w) s += lhist[w * 3 * NSEG + i];
        atomicAdd(&ws[i], s);
    }
}

// Single wave32. Exact f32 reduction of the 96 |mean diffs| using the f32 WMMA.
__global__ __launch_bounds__(32)
void ccl_finalize(const float* __restrict__ ws, float* __restrict__ out) {
    __shared__ float sdiff[NSEG];
    const int lane = threadIdx.x;

#pragma unroll
    for (int k = 0; k < 3; ++k) {
        const int i   = lane + 32 * k;
        const float c = ws[2 * NSEG + i];
        const float mp = (c > 0.0f) ? ws[i] / c : 0.0f;
        const float mt = (c > 0.0f) ? ws[NSEG + i] / c : 0.0f;
        sdiff[i] = fabsf(mp - mt);
    }
    __syncthreads();

    float total;
#if defined(__HIP_DEVICE_COMPILE__) && __has_builtin(__builtin_amdgcn_wmma_f32_16x16x4_f32)
    // A-matrix 16x4 f32 layout (ISA 7.12.2): lane L<16 holds A[M=L][K=0..1],
    // lane L>=16 holds A[M=L-16][K=2..3]. B = all-ones (layout-invariant).
    const int m     = lane & 15;
    const int khalf = lane >> 4;     // 0 -> K{0,1}, 1 -> K{2,3}

    v2f a1, a2, bones;
    // chunk 1: A[m][k] = sdiff[m*4 + k]            (values 0..63)
    a1.x = sdiff[(m << 2) + (khalf << 1) + 0];
    a1.y = sdiff[(m << 2) + (khalf << 1) + 1];
    // chunk 2: A[m][k] = sdiff[64 + m*2 + k], k<2  (values 64..95, rest 0)
    a2.x = (khalf == 0) ? sdiff[64 + (m << 1) + 0] : 0.0f;
    a2.y = (khalf == 0) ? sdiff[64 + (m << 1) + 1] : 0.0f;
    bones.x = 1.0f; bones.y = 1.0f;

    v8f cacc = {};
    // D[i][j] = sum_k A[i][k]  (row sums replicated over all 16 columns)
    cacc = __builtin_amdgcn_wmma_f32_16x16x4_f32(false, a1, false, bones,
                                                 (short)0, cacc, false, false);
    cacc = __builtin_amdgcn_wmma_f32_16x16x4_f32(false, a2, false, bones,
                                                 (short)0, cacc, false, false);
    // Column N=lane: lanes 0..15 hold M=0..7 in c[0..7], lanes 16..31 M=8..15.
    float partial = cacc[0] + cacc[1] + cacc[2] + cacc[3]
                  + cacc[4] + cacc[5] + cacc[6] + cacc[7];
    total = partial + __shfl_xor(partial, 16, 32);
#else
    float partial = sdiff[lane] + sdiff[lane + 32] + sdiff[lane + 64];
    for (int off = 16; off > 0; off >>= 1) partial += __shfl_xor(partial, off, 32);
    total = partial;
#endif

    if (lane == 0) out[0] = total * (1.0f / 96.0f);
}

extern "C" void kernel_launch(void* const* d_in, const int* in_sizes, int n_in,
                              void* d_out, int out_size, void* d_ws, size_t ws_size,
                              hipStream_t stream) {
    const float* pred = (const float*)d_in[0];
    const float* targ = (const float*)d_in[1];
    const float* img  = (const float*)d_in[2];
    float* ws  = (float*)d_ws;
    float* out = (float*)d_out;

    const int n    = in_sizes[0];   // 16*3*512*512 = 12,582,912
    const int nvec = n >> 2;        // float4 count (planes are 4-divisible)

    ccl_zero_ws<<<1, 128, 0, stream>>>(ws);

    int blocks = (nvec + 255) / 256;
    if (blocks > 1024) blocks = 1024;   // 8192 waves, ~12 float4 iters/thread
    ccl_hist<<<blocks, 256, 0, stream>>>((const f4*)pred, (const f4*)targ,
                                         (const f4*)img, ws, nvec);

    ccl_finalize<<<1, 32, 0, stream>>>(ws, out);
}